// EdgeEmbedding_11038065951284
// MI455X (gfx1250) — compile-verified
//
#include <hip/hip_runtime.h>

typedef float v2f __attribute__((ext_vector_type(2)));
typedef float v4f __attribute__((ext_vector_type(4)));
typedef float v8f __attribute__((ext_vector_type(8)));

#define CHANNELS   32
#define MAX_SCALAR 10
#define MAX_BASIS  30
#define N_TYPES    4
#define N_PAIRS    16              // N_TYPES * N_TYPES
#define ROW_F      320             // CHANNELS * MAX_SCALAR floats per table row
#define PR         336             // b64 stride per K-pair row: 2*336 % 64 == 32 -> conflict-free
#define OUT_ROW    1920            // MAX_BASIS * 2 * CHANNELS floats per edge

// Table in LDS, K-pair interleaved: element (k, col) lives at
//   ((k>>1)*PR + col) as a v2f, component (k&1).
// So the B operand chunk {row k0, row k0+1} at column col is ONE ds_load_b64.

template <bool FULL>
__device__ __forceinline__ void process_group(
    const v2f* __restrict__ tab,   // LDS, [8*PR] v2f
    float*     __restrict__ out,
    int eg, int E, int lane, int m, int hi,
    const int idx[2], const int sd[2])
{
    // Per-lane store base: rows (eg + 8*hi + r), column m.
    float* __restrict__ dp = out + (size_t)(eg + 8 * hi) * OUT_ROW + m;

    // ---- data rows p = 0..9 : exact one-hot f32 WMMA gather ----
    for (int p = 0; p < MAX_SCALAR; ++p) {
        // B chunks (side-independent). lane N = m; v2f = rows {4c4+2hi, 4c4+2hi+1}.
        v2f B[2][4];
        #pragma unroll
        for (int ct = 0; ct < 2; ++ct) {
            const int col = p * CHANNELS + ct * 16 + m;
            #pragma unroll
            for (int c4 = 0; c4 < 4; ++c4) {
                B[ct][c4] = tab[(2 * c4 + hi) * PR + col];
            }
        }

        float* __restrict__ dpp = dp + p * 64;

        #pragma unroll
        for (int side = 0; side < 2; ++side) {
            const float on = (p < sd[side]) ? 1.0f : 0.0f;  // mask folded into one-hot
            const int target = idx[side];

            // A (16x4 per chunk): lane M = m; VGPR0 K = 4*c4+2*hi, VGPR1 = +1.
            v2f A[4];
            #pragma unroll
            for (int c4 = 0; c4 < 4; ++c4) {
                const int k0 = 4 * c4 + 2 * hi;
                A[c4].x = (target == k0    ) ? on : 0.0f;
                A[c4].y = (target == k0 + 1) ? on : 0.0f;
            }

            #pragma unroll
            for (int ct = 0; ct < 2; ++ct) {
                v8f acc = {};  // C = 0
                #pragma unroll
                for (int c4 = 0; c4 < 4; ++c4) {
                    acc = __builtin_amdgcn_wmma_f32_16x16x4_f32(
                              false, A[c4], false, B[ct][c4],
                              (short)0, acc, false, false);
                }
                // D layout: VGPR r, lane l -> row (r + 8*hi), col (l & 15).
                #pragma unroll
                for (int r = 0; r < 8; ++r) {
                    if (FULL || (eg + 8 * hi + r) < E) {
                        dpp[(size_t)r * OUT_ROW + side * 32 + ct * 16] = acc[r];
                    }
                }
            }
        }
    }

    // ---- pad rows p = 10..29 : 1280 contiguous zero floats per edge ----
    const v4f z = {0.0f, 0.0f, 0.0f, 0.0f};
    float* __restrict__ zb = out + (size_t)eg * OUT_ROW + MAX_SCALAR * 2 * CHANNELS
                                 + lane * 4;
    for (int mm = 0; mm < 16; ++mm) {
        if (!FULL && (eg + mm >= E)) break;
        float* __restrict__ zm = zb + (size_t)mm * OUT_ROW;
        #pragma unroll
        for (int it = 0; it < 10; ++it) {
            *(v4f*)(zm + it * 128) = z;   // coalesced b128, immediate offsets
        }
    }
}

__global__ __launch_bounds__(256) void edge_emb_wmma_kernel(
    const float* __restrict__ emb_weight,     // [16, 320]
    const int*   __restrict__ atomic_numbers, // [N_NODES]
    const int*   __restrict__ edge_index,     // [2, E]
    const int*   __restrict__ a2i,            // [9]
    const int*   __restrict__ scalar_dims,    // [4]
    float*       __restrict__ out,            // [E, 30, 64]
    int E)
{
    __shared__ v2f tab[8 * PR];               // 21504 B, K-pair interleaved
    __shared__ int s_a2i[9];
    __shared__ int s_sd[4];

    // Cooperative load + reshuffle of the 16x320 table into LDS.
    {
        float* tf = (float*)tab;
        for (int i = threadIdx.x; i < N_PAIRS * ROW_F; i += blockDim.x) {
            const int k   = i / ROW_F;
            const int col = i - k * ROW_F;
            tf[((k >> 1) * PR + col) * 2 + (k & 1)] = emb_weight[i];
        }
    }
    if (threadIdx.x < 9) s_a2i[threadIdx.x] = a2i[threadIdx.x];
    if (threadIdx.x < 4) s_sd[threadIdx.x]  = scalar_dims[threadIdx.x];
    __syncthreads();

    const int lane = threadIdx.x & 31;
    const int wave = threadIdx.x >> 5;
    const int wavesPerBlock = blockDim.x >> 5;
    const int totalWaves = gridDim.x * wavesPerBlock;
    const int numGroups = (E + 15) >> 4;

    const int m  = lane & 15;   // edge-in-group (A's M row; also D column index)
    const int hi = lane >> 4;   // lane-half selector for K striping

    for (int g = blockIdx.x * wavesPerBlock + wave; g < numGroups; g += totalWaves) {
        const int eg = g << 4;
        const bool full = (eg + 16 <= E);            // wave-uniform
        // Lane-uniform activity: clamp so EXEC stays all-ones for WMMA.
        const int e = full ? (eg + m) : min(eg + m, E - 1);

        const int na = edge_index[e];
        const int nb = edge_index[E + e];
        const int ta = s_a2i[atomic_numbers[na]];
        const int tb = s_a2i[atomic_numbers[nb]];
        int idx[2];
        int sd[2];
        idx[0] = ta * N_TYPES + tb;  sd[0] = s_sd[ta];
        idx[1] = tb * N_TYPES + ta;  sd[1] = s_sd[tb];

        if (full) {
            process_group<true >(tab, out, eg, E, lane, m, hi, idx, sd);
        } else {
            process_group<false>(tab, out, eg, E, lane, m, hi, idx, sd);
        }
    }
}

extern "C" void kernel_launch(void* const* d_in, const int* in_sizes, int n_in,
                              void* d_out, int out_size, void* d_ws, size_t ws_size,
                              hipStream_t stream) {
    const float* emb  = (const float*)d_in[0];
    const int*   atn  = (const int*)d_in[1];
    const int*   ei   = (const int*)d_in[2];
    const int*   a2i  = (const int*)d_in[3];
    const int*   sdim = (const int*)d_in[4];
    float*       out  = (float*)d_out;

    const int E = in_sizes[2] / 2;
    const int numGroups = (E + 15) / 16;
    const int wavesPerBlock = 8;               // 256 threads
    int blocks = (numGroups + wavesPerBlock - 1) / wavesPerBlock;
    if (blocks > 32768) blocks = 32768;
    if (blocks < 1) blocks = 1;

    edge_emb_wmma_kernel<<<blocks, 256, 0, stream>>>(emb, atn, ei, a2i, sdim, out, E);

    // Second tuple output: edge_index passthrough, appended flat after the features.
    const long long mainElems = (long long)E * OUT_ROW;
    if ((long long)out_size > mainElems) {
        long long tail = (long long)out_size - mainElems;
        const long long cap = 2LL * E;
        if (tail > cap) tail = cap;
        hipMemcpyAsync((char*)d_out + mainElems * 4, d_in[2], (size_t)tail * 4,
                       hipMemcpyDeviceToDevice, stream);
    }
}